// NGNN_GCNConv_72541997629997
// MI455X (gfx1250) — compile-verified
//
#include <hip/hip_runtime.h>

typedef __attribute__((ext_vector_type(2))) float v2f;
typedef __attribute__((ext_vector_type(8))) float v8f;

// ---------------- elementwise helpers ----------------

__global__ void fill_zero_kernel(float* __restrict__ p, long long n) {
    long long i = (long long)blockIdx.x * blockDim.x + threadIdx.x;
    long long stride = (long long)gridDim.x * blockDim.x;
    for (; i < n; i += stride) p[i] = 0.0f;
}

// deg[dst[e]] += ew[e]  over all E edges (lane-coalesced index loads)
__global__ void deg_kernel(const long long* __restrict__ dst,
                           const float* __restrict__ ew,
                           float* __restrict__ deg, long long E) {
    long long i = (long long)blockIdx.x * blockDim.x + threadIdx.x;
    long long stride = (long long)gridDim.x * blockDim.x;
    for (; i < E; i += stride) {
        atomicAdd(&deg[(int)dst[i]], ew[i]);
    }
}

// deg -> dinv in place (self-loop weight 1 added here)
__global__ void dinv_kernel(float* __restrict__ deg, int n) {
    int i = blockIdx.x * blockDim.x + threadIdx.x;
    if (i < n) {
        float d = deg[i] + 1.0f; // self-loop
        deg[i] = (d > 0.0f) ? rsqrtf(d) : 0.0f;
    }
}

// ---------------- WMMA f32 GEMM: out[rows,128] = act(in[rows,128] @ W[128,128]^T + bias) ----------------
// W is staged in LDS (row stride 132 floats -> conflict-free v2f fragment reads:
// bank = (4*lm + k + kh) mod 64 is distinct for all 32 lanes, each b64 covering 2 banks).
// One wave computes a 16-row x 128-col strip with V_WMMA_F32_16X16X4_F32.
//   A frag (16x4):  lane L (M = L&15) holds in[row0+M][k + 2*(L>>4) .. +1]
//   B frag (4x16):  lane L (N = L&15) holds W[col0+N][k + 2*(L>>4) .. +1]  (B = W^T tile)
//   D (16x16): vgpr r, lane L -> row (L<16 ? r : r+8), col L&15
#define WSTRIDE 132

template <bool RELU, bool BIAS>
__global__ __launch_bounds__(256) void gemm128_wmma(const float* __restrict__ A,
                                                    const float* __restrict__ W,
                                                    const float* __restrict__ bias,
                                                    float* __restrict__ out, int rows) {
    __shared__ float Ws[128 * WSTRIDE];

    // cooperative stage of W into LDS (4096 float4 = 64 KB)
    for (int i = threadIdx.x; i < 4096; i += 256) {
        const int r = i >> 5;            // 32 float4 per 128-float row
        const int c = (i & 31) << 2;     // float column
        *(float4*)&Ws[r * WSTRIDE + c] = *(const float4*)&W[r * 128 + c];
    }
    __syncthreads();

    const int wave = (int)((blockIdx.x * blockDim.x + threadIdx.x) >> 5);
    const int lane = threadIdx.x & 31;
    const int row0 = wave * 16;
    if (row0 >= rows) return;

    const int lm = lane & 15;          // M (A) / N (B, D)
    const int kh = (lane >> 4) * 2;    // k sub-offset: 0 or 2
    const int mofs = (lane >> 4) * 8;  // D row offset: 0 or 8

    v8f acc[8] = {};
    const float* arow = A + (long long)(row0 + lm) * 128 + kh;
    const float* brow = Ws + lm * WSTRIDE + kh;

#pragma unroll 4
    for (int k = 0; k < 128; k += 4) {
        const v2f a = *(const v2f*)(arow + k);
        v2f b[8];
#pragma unroll
        for (int c = 0; c < 8; ++c)
            b[c] = *(const v2f*)(brow + c * (16 * WSTRIDE) + k);
#pragma unroll
        for (int c = 0; c < 8; ++c)
            acc[c] = __builtin_amdgcn_wmma_f32_16x16x4_f32(
                false, a, false, b[c], (short)0, acc[c], false, false);
    }

#pragma unroll
    for (int c = 0; c < 8; ++c) {
        const int col = c * 16 + lm;
        const float bv = BIAS ? bias[col] : 0.0f;
#pragma unroll
        for (int r = 0; r < 8; ++r) {
            float v = acc[c][r] + bv;
            if (RELU) v = fmaxf(v, 0.0f);
            out[(long long)(row0 + mofs + r) * 128 + col] = v;
        }
    }
}

// ---------------- edge scatter-add (dominant phase) ----------------
// Each wave handles batches of 32 edges: lane L loads edge base+L's metadata
// (coalesced), then the wave broadcasts (s, d, norm) per edge via v_readlane
// (SGPR-uniform base -> saddr-form b128 gather + 4 f32 atomics per lane).
__global__ void edge_agg_kernel(const long long* __restrict__ src,
                                const long long* __restrict__ dst,
                                const float* __restrict__ ew,
                                const float* __restrict__ dinv,
                                const float* __restrict__ h,
                                float* __restrict__ agg, long long E) {
    const int lane = threadIdx.x & 31;
    const long long wave = (long long)blockIdx.x * (blockDim.x >> 5) + (threadIdx.x >> 5);
    const long long nwaves = (long long)gridDim.x * (blockDim.x >> 5);

    for (long long base = wave * 32; base < E; base += nwaves * 32) {
        const int nb = (int)((E - base < 32) ? (E - base) : 32);
        int myS = 0, myD = 0;
        float myN = 0.0f;
        if (lane < nb) {
            myS = (int)src[base + lane];
            myD = (int)dst[base + lane];
            myN = dinv[myS] * ew[base + lane] * dinv[myD];
        }
        // warm L1/L2 for the 32 upcoming h rows (global_prefetch_b8)
        __builtin_prefetch(h + (long long)myS * 128, 0, 1);

        for (int j = 0; j < nb; ++j) {
            const int s = __builtin_amdgcn_readlane(myS, j);
            const int d = __builtin_amdgcn_readlane(myD, j);
            const float norm =
                __int_as_float(__builtin_amdgcn_readlane(__float_as_int(myN), j));
            const float4 v = ((const float4*)(h + (long long)s * 128))[lane];
            float* ap = agg + (long long)d * 128 + lane * 4;
            atomicAdd(ap + 0, v.x * norm);
            atomicAdd(ap + 1, v.y * norm);
            atomicAdd(ap + 2, v.z * norm);
            atomicAdd(ap + 3, v.w * norm);
        }
    }
}

// ---------------- conv epilogue: self-loop + bias + relu, in place on agg ----------------
__global__ void epilogue_kernel(const float* __restrict__ h,
                                const float* __restrict__ dinv,
                                const float* __restrict__ b_conv,
                                float* __restrict__ agg, long long total4) {
    const long long i4 = (long long)blockIdx.x * blockDim.x + threadIdx.x;
    if (i4 >= total4) return;
    const long long node = i4 >> 5;        // 32 float4 per node
    const int c = (int)(i4 & 31) << 2;     // channel base
    const float di = dinv[node];
    const float di2 = di * di;
    const float4 a = ((const float4*)agg)[i4];
    const float4 hv = ((const float4*)h)[i4];
    const float4 bv = *(const float4*)&b_conv[c];
    float4 r;
    r.x = fmaxf(a.x + hv.x * di2 + bv.x, 0.0f);
    r.y = fmaxf(a.y + hv.y * di2 + bv.y, 0.0f);
    r.z = fmaxf(a.z + hv.z * di2 + bv.z, 0.0f);
    r.w = fmaxf(a.w + hv.w * di2 + bv.w, 0.0f);
    ((float4*)agg)[i4] = r;
}

// ---------------- launch ----------------

extern "C" void kernel_launch(void* const* d_in, const int* in_sizes, int n_in,
                              void* d_out, int out_size, void* d_ws, size_t ws_size,
                              hipStream_t stream) {
    const float* x = (const float*)d_in[0];
    const long long* edge_index = (const long long*)d_in[1];
    const float* edge_weight = (const float*)d_in[2];
    const float* W_conv = (const float*)d_in[3];
    const float* b_conv = (const float*)d_in[4];
    const float* W_fc = (const float*)d_in[5];
    const float* b_fc = (const float*)d_in[6];
    const float* W_fc2 = (const float*)d_in[7];
    const float* b_fc2 = (const float*)d_in[8];

    const int N = in_sizes[0] / 128;
    const long long E = (long long)in_sizes[2];
    const long long NC = (long long)N * 128;

    const long long* src = edge_index;      // edge_index[0,:]
    const long long* dst = edge_index + E;  // edge_index[1,:]

    // workspace: deg/dinv [N] | h [N*128] | agg [N*128]
    float* deg = (float*)d_ws;
    float* h = deg + N;
    float* agg = h + NC;

    // 1. zero deg and agg
    fill_zero_kernel<<<512, 256, 0, stream>>>(deg, (long long)N);
    fill_zero_kernel<<<4096, 256, 0, stream>>>(agg, NC);

    // 2. degree accumulation (atomics over E)
    deg_kernel<<<4096, 256, 0, stream>>>(dst, edge_weight, deg, E);

    // 3. deg -> dinv (adds self-loop weight)
    dinv_kernel<<<(N + 255) / 256, 256, 0, stream>>>(deg, N);

    // 4. h = x @ W_conv^T  (WMMA f32, no bias/act)
    const int tiles = N / 16;             // N = 100000 -> 6250 full tiles
    const int gblocks = (tiles + 7) / 8;  // 8 waves (256 threads) per block
    gemm128_wmma<false, false><<<gblocks, 256, 0, stream>>>(x, W_conv, nullptr, h, N);

    // 5. edge scatter-add: agg[dst] += h[src] * norm
    edge_agg_kernel<<<4096, 256, 0, stream>>>(src, dst, edge_weight, deg, h, agg, E);

    // 6. agg = relu(agg + h*dinv^2 + b_conv)
    epilogue_kernel<<<(int)((NC / 4 + 255) / 256), 256, 0, stream>>>(h, deg, b_conv, agg, NC / 4);

    // 7. h = relu(agg @ W_fc^T + b_fc)   (reuse h buffer)
    gemm128_wmma<true, true><<<gblocks, 256, 0, stream>>>(agg, W_fc, b_fc, h, N);

    // 8. out = h @ W_fc2^T + b_fc2
    gemm128_wmma<false, true><<<gblocks, 256, 0, stream>>>(h, W_fc2, b_fc2, (float*)d_out, N);
}